// Encoder_19026705121764
// MI455X (gfx1250) — compile-verified
//
#include <hip/hip_runtime.h>
#include <hip/hip_bf16.h>

typedef __attribute__((ext_vector_type(2))) float v2f;
typedef __attribute__((ext_vector_type(8))) float v8f;

#define NH 256

// ---------------------------------------------------------------------------
// Utility: zero-fill (float4 granularity; all our buffers are /4 divisible)
// ---------------------------------------------------------------------------
__global__ __launch_bounds__(256)
void fill_zero_kernel(float4* __restrict__ p, int n4) {
    int i = blockIdx.x * 256 + threadIdx.x;
    if (i < n4) p[i] = make_float4(0.f, 0.f, 0.f, 0.f);
}

// ---------------------------------------------------------------------------
// Degree accumulation: out_deg[src[e]] += 1 ; in_deg[dst[e]] += 1
// ---------------------------------------------------------------------------
__global__ __launch_bounds__(256)
void deg_kernel(const int* __restrict__ src, const int* __restrict__ dst,
                float* __restrict__ out_deg, float* __restrict__ in_deg,
                int n_edges) {
    int i = blockIdx.x * 256 + threadIdx.x;
    if (i < n_edges) {
        atomicAdd(&out_deg[src[i]], 1.0f);
        atomicAdd(&in_deg[dst[i]], 1.0f);
    }
}

// d[i] = rsqrt(max(d[i], 1))   (in place over both norm vectors)
__global__ __launch_bounds__(256)
void norm_fin_kernel(float* __restrict__ d, int n) {
    int i = blockIdx.x * 256 + threadIdx.x;
    if (i < n) d[i] = __frsqrt_rn(fmaxf(d[i], 1.0f));
}

// ---------------------------------------------------------------------------
// GEMM via V_WMMA_F32_16X16X4_F32.
//   Y[N,256] = rowscale(X) @ W        (W is [kdim, 256] row-major)
// FUSE=true additionally applies the previous layer's epilogue to X on load:
//   x' = prelu(x * norm_dst[row] + b_in[k], a_in[k]) ; then * norm_src[row]
// Block = 128 threads = 4 waves; block computes a 16-row x 256-col tile,
// wave w computes cols [64w, 64w+64) via 4 accumulators.
// A frag (16x4 f32, 2 VGPR): lanes 0-15 -> K={k,k+1}, lanes 16-31 -> K={k+2,k+3}
// B frag (4x16 f32, 2 VGPR): same K split, N = lane&15
// D (16x16 f32, 8 VGPR):     M = r + 8*(lane>>4), N = lane&15
// ---------------------------------------------------------------------------
template<bool FUSE>
__global__ __launch_bounds__(128)
void gemm_wmma_kernel(const float* __restrict__ X, int kdim,
                      const float* __restrict__ W,
                      const float* __restrict__ norm_src,
                      const float* __restrict__ norm_dst,
                      const float* __restrict__ b_in,
                      const float* __restrict__ a_in,
                      float* __restrict__ Y) {
    const int lane = threadIdx.x & 31;
    const int wv   = threadIdx.x >> 5;
    const int row0 = blockIdx.x << 4;      // 16 rows per block
    const int mrow = lane & 15;
    const int hi   = lane >> 4;            // 0: K+{0,1}, 1: K+{2,3}
    const int colL = lane & 15;
    const int col0 = wv << 6;              // 64 cols per wave

    const int grow = row0 + mrow;
    const float ns = norm_src[grow];
    const float nd = FUSE ? norm_dst[grow] : 0.0f;
    const float* __restrict__ xrow = X + (size_t)grow * (size_t)kdim;

    v8f acc0 = {}, acc1 = {}, acc2 = {}, acc3 = {};

    #pragma unroll 2
    for (int k = 0; k < kdim; k += 4) {
        const int ka = k + (hi << 1);
        float x0 = xrow[ka];
        float x1 = xrow[ka + 1];
        if (FUSE) {
            float h0 = fmaf(x0, nd, b_in[ka]);
            float h1 = fmaf(x1, nd, b_in[ka + 1]);
            x0 = (h0 > 0.f) ? h0 : a_in[ka] * h0;
            x1 = (h1 > 0.f) ? h1 : a_in[ka + 1] * h1;
        }
        v2f a;
        a.x = x0 * ns;
        a.y = x1 * ns;

        const float* __restrict__ wp = W + (size_t)ka * NH + col0 + colL;
        v2f b0, b1, b2, b3;
        b0.x = wp[0];   b0.y = wp[NH + 0];
        b1.x = wp[16];  b1.y = wp[NH + 16];
        b2.x = wp[32];  b2.y = wp[NH + 32];
        b3.x = wp[48];  b3.y = wp[NH + 48];

        acc0 = __builtin_amdgcn_wmma_f32_16x16x4_f32(false, a, false, b0, (short)0, acc0, false, false);
        acc1 = __builtin_amdgcn_wmma_f32_16x16x4_f32(false, a, false, b1, (short)0, acc1, false, false);
        acc2 = __builtin_amdgcn_wmma_f32_16x16x4_f32(false, a, false, b2, (short)0, acc2, false, false);
        acc3 = __builtin_amdgcn_wmma_f32_16x16x4_f32(false, a, false, b3, (short)0, acc3, false, false);
    }

    float* __restrict__ yp = Y + (size_t)row0 * NH + col0 + colL;
    #pragma unroll
    for (int r = 0; r < 8; ++r) {
        const size_t moff = (size_t)(r + (hi << 3)) * NH;
        yp[moff + 0]  = acc0[r];
        yp[moff + 16] = acc1[r];
        yp[moff + 32] = acc2[r];
        yp[moff + 48] = acc3[r];
    }
}

// ---------------------------------------------------------------------------
// SpMM scatter: AGG[dst[e], :] += H[src[e], :]   (one wave per edge)
// H/AGG fit in the 192MB L2, so both gather loads and the f32 atomics
// stay on-chip. Coalesced float4 loads + per-lane global_atomic_add_f32.
// ---------------------------------------------------------------------------
__global__ __launch_bounds__(256)
void spmm_kernel(const float* __restrict__ H, const int* __restrict__ src,
                 const int* __restrict__ dst, float* __restrict__ AGG,
                 int n_edges) {
    int gtid = blockIdx.x * 256 + threadIdx.x;
    int e    = gtid >> 5;
    int lane = threadIdx.x & 31;
    if (e >= n_edges) return;
    int s = __builtin_amdgcn_readfirstlane(src[e]);
    int d = __builtin_amdgcn_readfirstlane(dst[e]);
    const float4* __restrict__ hs = (const float4*)(H + (size_t)s * NH);
    float* __restrict__ ad = AGG + (size_t)d * NH;
    #pragma unroll
    for (int j = 0; j < 2; ++j) {
        float4 v = hs[lane + 32 * j];
        int base = (lane + 32 * j) * 4;
        atomicAdd(ad + base + 0, v.x);
        atomicAdd(ad + base + 1, v.y);
        atomicAdd(ad + base + 2, v.z);
        atomicAdd(ad + base + 3, v.w);
    }
}

// ---------------------------------------------------------------------------
// Final epilogue: out = prelu(AGG * norm_dst[row] + b2[col], a2[col])
// ---------------------------------------------------------------------------
__global__ __launch_bounds__(256)
void epilogue_kernel(const float* __restrict__ AGG,
                     const float* __restrict__ norm_dst,
                     const float* __restrict__ b, const float* __restrict__ a,
                     float* __restrict__ out, int total) {
    int i = (blockIdx.x * 256 + threadIdx.x) * 4;
    if (i >= total) return;
    int row = i >> 8;       // NH == 256
    int col = i & (NH - 1);
    float nd = norm_dst[row];
    float4 v = *(const float4*)(AGG + i);
    float4 bb = *(const float4*)(b + col);
    float4 aa = *(const float4*)(a + col);
    float h0 = fmaf(v.x, nd, bb.x);
    float h1 = fmaf(v.y, nd, bb.y);
    float h2 = fmaf(v.z, nd, bb.z);
    float h3 = fmaf(v.w, nd, bb.w);
    float4 r;
    r.x = (h0 > 0.f) ? h0 : aa.x * h0;
    r.y = (h1 > 0.f) ? h1 : aa.y * h1;
    r.z = (h2 > 0.f) ? h2 : aa.z * h2;
    r.w = (h3 > 0.f) ? h3 : aa.w * h3;
    *(float4*)(out + i) = r;
}

// ---------------------------------------------------------------------------
extern "C" void kernel_launch(void* const* d_in, const int* in_sizes, int n_in,
                              void* d_out, int out_size, void* d_ws, size_t ws_size,
                              hipStream_t stream) {
    const float* features = (const float*)d_in[0];
    const int*   src      = (const int*)d_in[1];
    const int*   dst      = (const int*)d_in[2];
    const float* W1       = (const float*)d_in[3];
    const float* b1       = (const float*)d_in[4];
    const float* a1       = (const float*)d_in[5];
    const float* W2       = (const float*)d_in[6];
    const float* b2       = (const float*)d_in[7];
    const float* a2       = (const float*)d_in[8];

    const int n_h     = in_sizes[4];                  // 256
    const int n_in_f  = in_sizes[3] / n_h;            // 512
    const int n_nodes = in_sizes[0] / n_in_f;         // 100000
    const int n_edges = in_sizes[1];                  // 3200000
    (void)n_h;

    // Workspace layout (floats): [norm_src N][norm_dst N][H N*256][AGG N*256]
    float* ws    = (float*)d_ws;
    float* nsrc  = ws;
    float* ndst  = ws + n_nodes;
    float* H     = ndst + n_nodes;
    float* AGG   = H + (size_t)n_nodes * NH;

    const int feat_total = n_nodes * NH;              // 25.6M floats

    // 1) degree norms (zero counters, accumulate, rsqrt)
    {
        int n4 = (2 * n_nodes) / 4;
        fill_zero_kernel<<<(n4 + 255) / 256, 256, 0, stream>>>((float4*)nsrc, n4);
        deg_kernel<<<(n_edges + 255) / 256, 256, 0, stream>>>(src, dst, nsrc, ndst, n_edges);
        norm_fin_kernel<<<(2 * n_nodes + 255) / 256, 256, 0, stream>>>(nsrc, 2 * n_nodes);
    }

    // 2) zero AGG, GEMM1 -> H, SpMM -> AGG
    {
        int n4 = feat_total / 4;
        fill_zero_kernel<<<(n4 + 255) / 256, 256, 0, stream>>>((float4*)AGG, n4);
    }
    gemm_wmma_kernel<false><<<n_nodes / 16, 128, 0, stream>>>(
        features, n_in_f, W1, nsrc, nullptr, nullptr, nullptr, H);
    {
        long long threads = (long long)n_edges * 32;
        spmm_kernel<<<(int)((threads + 255) / 256), 256, 0, stream>>>(H, src, dst, AGG, n_edges);
    }

    // 3) GEMM2 reads AGG with layer-1 epilogue fused into the A-load -> H
    gemm_wmma_kernel<true><<<n_nodes / 16, 128, 0, stream>>>(
        AGG, NH, W2, nsrc, ndst, b1, a1, H);

    // 4) zero AGG (after GEMM2 consumed it), SpMM -> AGG
    {
        int n4 = feat_total / 4;
        fill_zero_kernel<<<(n4 + 255) / 256, 256, 0, stream>>>((float4*)AGG, n4);
    }
    {
        long long threads = (long long)n_edges * 32;
        spmm_kernel<<<(int)((threads + 255) / 256), 256, 0, stream>>>(H, src, dst, AGG, n_edges);
    }

    // 5) final epilogue -> d_out
    epilogue_kernel<<<(feat_total / 4 + 255) / 256, 256, 0, stream>>>(
        AGG, ndst, b2, a2, (float*)d_out, feat_total);
}